// LightGCN_15934328668924
// MI455X (gfx1250) — compile-verified
//
#include <hip/hip_runtime.h>
#include <hip/hip_bf16.h>

#define NUM_USERS 100000
#define NUM_ITEMS 50000
#define N_NODES   (NUM_USERS + NUM_ITEMS)
#define LATENT    64
#define NNZ_E     4000000
#define BATCH_SZ  16384

typedef __attribute__((ext_vector_type(2))) float v2f;
typedef __attribute__((ext_vector_type(8))) float v8f;

// -------- init: X = acc = concat(user_table, item_table), float4 vectorized --------
__global__ void lgcn_init(const float* __restrict__ utab, const float* __restrict__ itab,
                          float* __restrict__ X, float* __restrict__ acc) {
    long i4 = (long)blockIdx.x * blockDim.x + threadIdx.x;        // float4 index
    long n4 = (long)N_NODES * LATENT / 4;
    if (i4 >= n4) return;
    long i = i4 * 4;
    const long ub = (long)NUM_USERS * LATENT;
    float4 v;
    if (i < ub) v = *(const float4*)(utab + i);
    else        v = *(const float4*)(itab + (i - ub));
    *(float4*)(X + i)   = v;
    *(float4*)(acc + i) = v;
}

// -------- zero Y --------
__global__ void lgcn_zero(float* __restrict__ Y) {
    long i4 = (long)blockIdx.x * blockDim.x + threadIdx.x;
    long n4 = (long)N_NODES * LATENT / 4;
    if (i4 >= n4) return;
    *(float4*)(Y + i4 * 4) = make_float4(0.f, 0.f, 0.f, 0.f);
}

// -------- acc += Y --------
__global__ void lgcn_accum(float* __restrict__ acc, const float* __restrict__ Y) {
    long i4 = (long)blockIdx.x * blockDim.x + threadIdx.x;
    long n4 = (long)N_NODES * LATENT / 4;
    if (i4 >= n4) return;
    float4 a = *(float4*)(acc + i4 * 4);
    float4 y = *(const float4*)(Y + i4 * 4);
    a.x += y.x; a.y += y.y; a.z += y.z; a.w += y.w;
    *(float4*)(acc + i4 * 4) = a;
}

// -------- SpMM: one wave per edge; lane moves float2 (32 lanes x 8B = 256B row) --------
// Y[dst] += val * X[src]. X/Y/acc (115MB) stay resident in the 192MB L2, so this
// is an L2 gather/atomic-rate problem: coalesced b64 gather + native f32 atomic
// scatter. Edge lists are streamed with NT hints so they don't evict the
// node-feature working set from L2 (HBM re-stream cost ~6us at 23.3 TB/s).
__global__ void lgcn_spmm(const int* __restrict__ esrc, const int* __restrict__ edst,
                          const float* __restrict__ evals,
                          const float* __restrict__ X, float* __restrict__ Y) {
    int tid  = blockIdx.x * blockDim.x + threadIdx.x;
    int edge = tid >> 5;
    int lane = threadIdx.x & 31;
    if (edge >= NNZ_E) return;
    int   s = __builtin_nontemporal_load(esrc + edge);
    int   d = __builtin_nontemporal_load(edst + edge);
    float v = __builtin_nontemporal_load(evals + edge);
    const float* xr = X + (long)s * LATENT + 2 * lane;
    float2 x2 = *(const float2*)xr;
    float* yp = Y + (long)d * LATENT + 2 * lane;
    // unsafeAtomicAdd guarantees lowering to native global_atomic_add_f32
    // (non-returning variant since the result is unused).
    unsafeAtomicAdd(yp,     x2.x * v);
    unsafeAtomicAdd(yp + 1, x2.y * v);
}

// -------- batched dot via V_WMMA_F32_16X16X4_F32 --------
// One wave handles 16 batch elements. A = U-tile (16x4), B = I-tile^T (4x16),
// accumulate 16 K-chunks of 4 into the 16x16 f32 C, then read the diagonal.
// Per ISA layout (16x4 f32 A): lane l -> row M=l%16, VGPR0/1 = K = 2*(l/16)+{0,1};
// B mirrors with N=l%16. C diag: lanes 0-7 hold M=0-7 in VGPR=lane,
// lanes 24-31 hold M=8-15 in VGPR=lane-24.
__global__ void lgcn_gamma_wmma(const float* __restrict__ acc,
                                const int* __restrict__ users,
                                const int* __restrict__ items,
                                float* __restrict__ out) {
    int wave = (blockIdx.x * blockDim.x + threadIdx.x) >> 5;   // tile id, 0..1023
    int lane = threadIdx.x & 31;
    int b    = wave * 16 + (lane & 15);
    int half = (lane >> 4) & 1;

    const float* urow = acc + (long)users[b] * LATENT;
    const float* irow = acc + ((long)NUM_USERS + items[b]) * LATENT;

    v8f c = {};
#pragma unroll
    for (int kk = 0; kk < 16; ++kk) {
        int off = 4 * kk + 2 * half;
        v2f a  = *(const v2f*)(urow + off);
        v2f bb = *(const v2f*)(irow + off);
        // 8 args: (neg_a, A, neg_b, B, c_mod, C, reuse_a, reuse_b)
        c = __builtin_amdgcn_wmma_f32_16x16x4_f32(false, a, false, bb,
                                                  (short)0, c, false, false);
    }

    // Diagonal extraction: row index in tile and which accumulator VGPR holds it.
    int r = ((lane < 8) ? lane : (lane - 24)) & 7;
    float v = c[0];
#pragma unroll
    for (int i = 1; i < 8; ++i) if (r == i) v = c[i];

    // gamma = dot(acc_u, acc_i) / (L+1)^2 = dot / 16
    if (lane < 8)        out[wave * 16 + lane]        = v * 0.0625f;
    else if (lane >= 24) out[wave * 16 + (lane - 16)] = v * 0.0625f;
}

extern "C" void kernel_launch(void* const* d_in, const int* in_sizes, int n_in,
                              void* d_out, int out_size, void* d_ws, size_t ws_size,
                              hipStream_t stream) {
    const int*   users = (const int*)d_in[0];
    const int*   items = (const int*)d_in[1];
    const int*   esrc  = (const int*)d_in[2];
    const int*   edst  = (const int*)d_in[3];
    const float* evals = (const float*)d_in[4];
    const float* utab  = (const float*)d_in[5];
    const float* itab  = (const float*)d_in[6];
    float* out = (float*)d_out;

    const long NE = (long)N_NODES * LATENT;       // 9,600,000 floats
    float* X   = (float*)d_ws;
    float* Y   = X + NE;
    float* acc = Y + NE;                          // total 115.2 MB of d_ws

    const long n4      = NE / 4;                  // 2,400,000 float4s
    const int  blk     = 256;
    const int  gridEl  = (int)((n4 + blk - 1) / blk);

    lgcn_init<<<gridEl, blk, 0, stream>>>(utab, itab, X, acc);

    for (int layer = 0; layer < 3; ++layer) {
        lgcn_zero<<<gridEl, blk, 0, stream>>>(Y);
        // one wave (32 lanes) per edge
        long spmmThreads = (long)NNZ_E * 32;
        lgcn_spmm<<<(int)(spmmThreads / blk), blk, 0, stream>>>(esrc, edst, evals, X, Y);
        lgcn_accum<<<gridEl, blk, 0, stream>>>(acc, Y);
        float* t = X; X = Y; Y = t;
    }

    // 1024 tiles of 16 batch elements; 8 waves per 256-thread block -> 128 blocks
    lgcn_gamma_wmma<<<BATCH_SZ / 16 / 8, blk, 0, stream>>>(acc, users, items, out);
}